// GPT2AttentionKIVI_26371099197454
// MI455X (gfx1250) — compile-verified
//
#include <hip/hip_runtime.h>
#include <math.h>

typedef __attribute__((ext_vector_type(2))) float v2f;
typedef __attribute__((ext_vector_type(8))) float v8f;

#define B_ 2
#define S_ 2048
#define E_ 1024
#define H_ 16
#define DH_ 64
#define QLEN_ 2016   // S - RESIDUAL_LEN
#define CLIP_ 10000.0f
#define EPS_ 1e-8f
#define QLEVELS_ 3.0f   // 2^2 - 1

__device__ __forceinline__ v8f wmma_f32(v2f a, v2f b, v8f c) {
  // D = A(16x4,f32) x B(4x16,f32) + C(16x16,f32)
  return __builtin_amdgcn_wmma_f32_16x16x4_f32(
      false, a, false, b, (short)0, c, false, false);
}

// ---------------------------------------------------------------------------
// QKV GEMM: X(4096x1024) @ W(1024x3072) + bias -> Q/K/V in (B,H,S,Dh).
// 8 waves/block; each wave computes a 16x64 strip = 4 independent WMMA
// accumulator chains, reusing one A fragment per K-step across all 4 tiles.
// ---------------------------------------------------------------------------
__global__ __launch_bounds__(256) void qkv_gemm_kernel(
    const float* __restrict__ X, const float* __restrict__ W,
    const float* __restrict__ bias, float* __restrict__ Qo,
    float* __restrict__ Ko, float* __restrict__ Vo) {
  const int lane = threadIdx.x & 31;
  const int wid  = threadIdx.x >> 5;
  const int half = lane >> 4;
  const int l16  = lane & 15;
  const int m0 = blockIdx.y * 16;                    // 0..4080
  const int n0 = (blockIdx.x * 8 + wid) * 64;        // 0..3008, 64-aligned
  const int K = E_, N = 3 * E_;

  const float* arow = X + (long)(m0 + l16) * K;

  v8f c0 = {}, c1 = {}, c2 = {}, c3 = {};
  for (int k = 0; k < K; k += 16) {
#pragma unroll
    for (int kk = 0; kk < 16; kk += 4) {
      const int kb = k + kk + 2 * half;
      v2f a = *(const v2f*)(arow + kb);
      const float* w0 = W + (long)kb * N + n0 + l16;
      const float* w1 = w0 + N;
      v2f b;
      b.x = w0[0];  b.y = w1[0];  c0 = wmma_f32(a, b, c0);
      b.x = w0[16]; b.y = w1[16]; c1 = wmma_f32(a, b, c1);
      b.x = w0[32]; b.y = w1[32]; c2 = wmma_f32(a, b, c2);
      b.x = w0[48]; b.y = w1[48]; c3 = wmma_f32(a, b, c3);
    }
  }

  // 64-aligned strip => single Q/K/V target and single head for all 4 tiles
  const int which = n0 >> 10;              // 0=Q 1=K 2=V
  const int h     = (n0 & 1023) >> 6;
  float* dst = (which == 0) ? Qo : (which == 1) ? Ko : Vo;
  v8f cc[4] = {c0, c1, c2, c3};
#pragma unroll
  for (int t = 0; t < 4; ++t) {
    const int dh = t * 16 + l16;
    const float bb = bias[n0 + dh];
#pragma unroll
    for (int i = 0; i < 8; ++i) {
      const int m = m0 + i + 8 * half;     // global row in (B*S)
      const int b = m >> 11;
      const int s = m & 2047;
      dst[(((long)(b * H_ + h) * S_ + s) * DH_) + dh] = cc[t][i] + bb;
    }
  }
}

// ---------------------------------------------------------------------------
// K fake-quant: per (b,h) min/max over first QLEN_ tokens x 64 channels
// (group g == Dh collapses "per-channel" stats to one scale per head).
// ---------------------------------------------------------------------------
__global__ __launch_bounds__(256) void kquant_kernel(float* __restrict__ Kv) {
  const int head = blockIdx.x;              // 0..B*H-1
  float* base = Kv + (long)head * S_ * DH_;
  const int n = QLEN_ * DH_;

  float mn = 3.402823466e38f, mx = -3.402823466e38f;
  for (int i = threadIdx.x; i < n; i += 256) {
    float x = fminf(fmaxf(base[i], -CLIP_), CLIP_);
    mn = fminf(mn, x);
    mx = fmaxf(mx, x);
  }
  __shared__ float smn[256], smx[256];
  smn[threadIdx.x] = mn; smx[threadIdx.x] = mx;
  __syncthreads();
  for (int s = 128; s > 0; s >>= 1) {
    if (threadIdx.x < s) {
      smn[threadIdx.x] = fminf(smn[threadIdx.x], smn[threadIdx.x + s]);
      smx[threadIdx.x] = fmaxf(smx[threadIdx.x], smx[threadIdx.x + s]);
    }
    __syncthreads();
  }
  const float gmn = smn[0];
  const float scale = fmaxf((smx[0] - gmn) / QLEVELS_, EPS_);
  for (int i = threadIdx.x; i < n; i += 256) {
    float x = fminf(fmaxf(base[i], -CLIP_), CLIP_);
    base[i] = rintf((x - gmn) / scale) * scale + gmn;
  }
}

// ---------------------------------------------------------------------------
// V fake-quant: per token, min/max over the 64 channels. One wave per token.
// ---------------------------------------------------------------------------
__global__ __launch_bounds__(256) void vquant_kernel(float* __restrict__ Vv) {
  const int lane = threadIdx.x & 31;
  const int wtok = blockIdx.x * 8 + (threadIdx.x >> 5);   // 0..B*H*QLEN-1
  const int head = wtok / QLEN_;
  const int s    = wtok % QLEN_;
  float* p = Vv + ((long)head * S_ + s) * DH_;

  float x0 = fminf(fmaxf(p[lane],      -CLIP_), CLIP_);
  float x1 = fminf(fmaxf(p[lane + 32], -CLIP_), CLIP_);
  float mn = fminf(x0, x1), mx = fmaxf(x0, x1);
#pragma unroll
  for (int m = 1; m < 32; m <<= 1) {
    mn = fminf(mn, __shfl_xor(mn, m, 32));
    mx = fmaxf(mx, __shfl_xor(mx, m, 32));
  }
  const float scale = fmaxf((mx - mn) / QLEVELS_, EPS_);
  p[lane]      = rintf((x0 - mn) / scale) * scale + mn;
  p[lane + 32] = rintf((x1 - mn) / scale) * scale + mn;
}

// ---------------------------------------------------------------------------
// Flash attention: one wave per 16-row q-tile; streaming softmax over 16-wide
// k tiles. Scores use two interleaved WMMA accumulator chains; PV uses four.
// Output written to Y in (B,S,E) layout ready for the projection GEMM.
// ---------------------------------------------------------------------------
__global__ __launch_bounds__(256) void attn_kernel(
    const float* __restrict__ Q, const float* __restrict__ Kv,
    const float* __restrict__ Vv, float* __restrict__ Y) {
  __shared__ float pbuf[8][256];
  const int lane = threadIdx.x & 31;
  const int wid  = threadIdx.x >> 5;
  const int half = lane >> 4;
  const int l16  = lane & 15;
  const int bh    = blockIdx.x;                 // 0..31
  const int qtile = blockIdx.y * 8 + wid;       // 0..127
  const int qbase = qtile * 16;

  const float* Qb = Q  + (long)bh * S_ * DH_;
  const float* Kb = Kv + (long)bh * S_ * DH_;
  const float* Vb = Vv + (long)bh * S_ * DH_;
  float* P = &pbuf[wid][0];

  // Preload A-fragments of the 16x64 Q tile (16 k-steps of 4).
  v2f aq[16];
#pragma unroll
  for (int t = 0; t < 16; ++t)
    aq[t] = *(const v2f*)(Qb + (long)(qbase + l16) * DH_ + t * 4 + 2 * half);

  v8f o0 = {}, o1 = {}, o2 = {}, o3 = {};
  float mrow[8], lrow[8];
#pragma unroll
  for (int i = 0; i < 8; ++i) { mrow[i] = -INFINITY; lrow[i] = 0.f; }

  for (int j = 0; j <= qtile; ++j) {
    const int jb = j * 16;
    // prefetch next k-tile of K and V while we compute on this one
    if (j < qtile) {
      __builtin_prefetch(Kb + (long)(jb + 16 + l16) * DH_, 0, 1);
      __builtin_prefetch(Vb + (long)(jb + 16 + l16) * DH_, 0, 1);
    }
    // ---- S = Q Ktile^T : B[kdim][n=kpos] = K[jb+n][kdim]; 2 indep chains
    v8f s0 = {}, s1 = {};
    const float* krow = Kb + (long)(jb + l16) * DH_ + 2 * half;
#pragma unroll
    for (int t = 0; t < 16; t += 2) {
      v2f b0 = *(const v2f*)(krow + t * 4);
      v2f b1 = *(const v2f*)(krow + t * 4 + 4);
      s0 = wmma_f32(aq[t], b0, s0);
      s1 = wmma_f32(aq[t + 1], b1, s1);
    }
    v8f sc = s0 + s1;
#pragma unroll
    for (int i = 0; i < 8; ++i) sc[i] *= 0.125f;   // 1/sqrt(64)
    if (j == qtile) {
      const int col = jb + l16;
#pragma unroll
      for (int i = 0; i < 8; ++i)
        if (col > qbase + i + 8 * half) sc[i] = -INFINITY;
    }
    // ---- streaming softmax (rows live per VGPR-index within a lane half)
    float fac[8];
#pragma unroll
    for (int i = 0; i < 8; ++i) {
      float rmax = sc[i];
      rmax = fmaxf(rmax, __shfl_xor(rmax, 1, 32));
      rmax = fmaxf(rmax, __shfl_xor(rmax, 2, 32));
      rmax = fmaxf(rmax, __shfl_xor(rmax, 4, 32));
      rmax = fmaxf(rmax, __shfl_xor(rmax, 8, 32));
      const float mnew = fmaxf(mrow[i], rmax);
      fac[i] = expf(mrow[i] - mnew);
      const float pv = expf(sc[i] - mnew);
      sc[i] = pv;
      float rsum = pv;
      rsum += __shfl_xor(rsum, 1, 32);
      rsum += __shfl_xor(rsum, 2, 32);
      rsum += __shfl_xor(rsum, 4, 32);
      rsum += __shfl_xor(rsum, 8, 32);
      lrow[i] = lrow[i] * fac[i] + rsum;
      mrow[i] = mnew;
    }
#pragma unroll
    for (int i = 0; i < 8; ++i) {
      o0[i] *= fac[i]; o1[i] *= fac[i]; o2[i] *= fac[i]; o3[i] *= fac[i];
    }
    // ---- transpose P from C-layout to A-layout through per-wave LDS
#pragma unroll
    for (int i = 0; i < 8; ++i) P[(i + 8 * half) * 16 + l16] = sc[i];
    asm volatile("s_wait_dscnt 0" ::: "memory");   // in-wave LDS RAW fence
    // ---- O += P @ Vtile : 4 independent accumulator chains
#pragma unroll
    for (int ks = 0; ks < 4; ++ks) {
      v2f ap = *(const v2f*)(P + l16 * 16 + ks * 4 + 2 * half);
      const float* v0 = Vb + (long)(jb + ks * 4 + 2 * half) * DH_ + l16;
      const float* v1 = v0 + DH_;
      v2f b;
      b.x = v0[0];  b.y = v1[0];  o0 = wmma_f32(ap, b, o0);
      b.x = v0[16]; b.y = v1[16]; o1 = wmma_f32(ap, b, o1);
      b.x = v0[32]; b.y = v1[32]; o2 = wmma_f32(ap, b, o2);
      b.x = v0[48]; b.y = v1[48]; o3 = wmma_f32(ap, b, o3);
    }
  }

  // ---- normalize + store to (B,S,E)
  const int b = bh >> 4, h = bh & 15;
#pragma unroll
  for (int i = 0; i < 8; ++i) {
    const float inv = 1.0f / lrow[i];
    const int row = qbase + i + 8 * half;
    const long base = ((long)b * S_ + row) * E_ + h * DH_;
    Y[base + l16]      = o0[i] * inv;
    Y[base + 16 + l16] = o1[i] * inv;
    Y[base + 32 + l16] = o2[i] * inv;
    Y[base + 48 + l16] = o3[i] * inv;
  }
}

// ---------------------------------------------------------------------------
// Output projection: Y(4096x1024) @ W_proj(1024x1024) + b_proj -> out.
// Same 16x64-per-wave structure as the QKV GEMM.
// ---------------------------------------------------------------------------
__global__ __launch_bounds__(256) void proj_gemm_kernel(
    const float* __restrict__ Yin, const float* __restrict__ W,
    const float* __restrict__ bias, float* __restrict__ Out) {
  const int lane = threadIdx.x & 31;
  const int wid  = threadIdx.x >> 5;
  const int half = lane >> 4;
  const int l16  = lane & 15;
  const int m0 = blockIdx.y * 16;
  const int n0 = (blockIdx.x * 8 + wid) * 64;
  const int K = E_, N = E_;

  const float* arow = Yin + (long)(m0 + l16) * K;

  v8f c0 = {}, c1 = {}, c2 = {}, c3 = {};
  for (int k = 0; k < K; k += 16) {
#pragma unroll
    for (int kk = 0; kk < 16; kk += 4) {
      const int kb = k + kk + 2 * half;
      v2f a = *(const v2f*)(arow + kb);
      const float* w0 = W + (long)kb * N + n0 + l16;
      const float* w1 = w0 + N;
      v2f b;
      b.x = w0[0];  b.y = w1[0];  c0 = wmma_f32(a, b, c0);
      b.x = w0[16]; b.y = w1[16]; c1 = wmma_f32(a, b, c1);
      b.x = w0[32]; b.y = w1[32]; c2 = wmma_f32(a, b, c2);
      b.x = w0[48]; b.y = w1[48]; c3 = wmma_f32(a, b, c3);
    }
  }
  v8f cc[4] = {c0, c1, c2, c3};
#pragma unroll
  for (int t = 0; t < 4; ++t) {
    const int ncol = n0 + t * 16 + l16;
    const float bb = bias[ncol];
#pragma unroll
    for (int i = 0; i < 8; ++i) {
      const int m = m0 + i + 8 * half;
      Out[(long)m * N + ncol] = cc[t][i] + bb;
    }
  }
}

// ---------------------------------------------------------------------------
extern "C" void kernel_launch(void* const* d_in, const int* in_sizes, int n_in,
                              void* d_out, int out_size, void* d_ws,
                              size_t ws_size, hipStream_t stream) {
  const float* x      = (const float*)d_in[0];
  const float* W_attn = (const float*)d_in[1];
  const float* b_attn = (const float*)d_in[2];
  const float* W_proj = (const float*)d_in[3];
  const float* b_proj = (const float*)d_in[4];
  float* out = (float*)d_out;

  float* ws = (float*)d_ws;
  const long PER = (long)B_ * H_ * S_ * DH_;  // 4 Mi floats
  float* Q = ws;
  float* K = Q + PER;
  float* V = K + PER;
  float* Y = V + PER;

  // 1) fused QKV projection + head split (WMMA f32, 16x64 strips)
  qkv_gemm_kernel<<<dim3(6, 256), 256, 0, stream>>>(x, W_attn, b_attn, Q, K, V);
  // 2) KIVI fake-quant of K (per head) and V (per token), residual kept fp32
  kquant_kernel<<<B_ * H_, 256, 0, stream>>>(K);
  vquant_kernel<<<(B_ * H_ * QLEN_) / 8, 256, 0, stream>>>(V);
  // 3) causal flash attention (WMMA f32)
  attn_kernel<<<dim3(B_ * H_, 16), 256, 0, stream>>>(Q, K, V, Y);
  // 4) output projection (WMMA f32, 16x64 strips)
  proj_gemm_kernel<<<dim3(2, 256), 256, 0, stream>>>(Y, W_proj, b_proj, out);
}